// Convolutional_Reduced_Rank_Regression_42992622633092
// MI455X (gfx1250) — compile-verified
//
#include <hip/hip_runtime.h>
#include <hip/hip_bf16.h>
#include <math.h>

// ---------------- problem constants ----------------
#define T_LEN   50000
#define W_LEN   101
#define N_IN    128
#define RANK    10
#define N_OUT   64

#define KPAD    104      // 101 taps padded to multiple of 4
#define KSTEPS  26       // KPAD / 4
#define ROWS    32       // filter rows: 0..9 real, 16..25 imag, rest zero

#define CH1     512      // time chunk per workgroup, pass 1
#define NSUB1   4        // subtiles of 16 t per wave (512 / (16*8 waves))
#define CH3     128      // time chunk per workgroup, pass 3 (1 subtile/wave)

typedef __attribute__((ext_vector_type(2))) float v2f;
typedef __attribute__((ext_vector_type(8))) float v8f;

// Compute one 16(t) x 32(filter-row) fp32 conv tile with V_WMMA_F32_16X16X4_F32.
// Kf: [ROWS][KPAD] flipped kernels in LDS.  Xw: zero-padded X window in LDS,
// origin at (T0 - 50).  xbase = t0 - T0 for this wave's 16-t subtile.
// acc_re: filter rows 0..15 (real k=0..9), acc_im: rows 16..31 (imag k=0..9).
// Lane L holds column t = t0 + (L&15); register r holds row m = (L>>4)*8 + r.
__device__ __forceinline__ void conv_tile(const float* __restrict__ Xw, int xbase,
                                          const float* __restrict__ Kf,
                                          int lane, v8f& acc_re, v8f& acc_im) {
    const int col = lane & 15;          // N index (time)
    const int m   = lane & 15;          // M index (filter row within tile)
    const int kb  = (lane >> 4) << 1;   // K base: lanes 0-15 -> taps {0,1}, 16-31 -> {2,3}
#pragma unroll
    for (int r = 0; r < 8; ++r) { acc_re[r] = 0.0f; acc_im[r] = 0.0f; }
    for (int s = 0; s < KSTEPS; ++s) {
        const int u0 = s << 2;
        v2f a0, a1, b;
        const float* k0 = &Kf[m * KPAD + u0 + kb];           // A rows 0..15
        a0.x = k0[0]; a0.y = k0[1];
        const float* k1 = &Kf[(m + 16) * KPAD + u0 + kb];    // A rows 16..31
        a1.x = k1[0]; a1.y = k1[1];
        const int bi = xbase + col + u0 + kb;                // B[kk][t] = X[t-50+u0+kk]
        b.x = Xw[bi]; b.y = Xw[bi + 1];
        acc_re = __builtin_amdgcn_wmma_f32_16x16x4_f32(
            false, a0, false, b, (short)0, acc_re, false, false);
        acc_im = __builtin_amdgcn_wmma_f32_16x16x4_f32(
            false, a1, false, b, (short)0, acc_im, false, false);
    }
}

// Build flipped, padded filter bank in LDS.
// c[t] = sum_u X[t-50+u] * Kc[100-u]  (u in [0,101)), rows 0..9 real, 16..25 imag.
__device__ __forceinline__ void stage_filters(const float* __restrict__ K,
                                              float* __restrict__ Kf, int tid) {
    for (int i = tid; i < ROWS * KPAD; i += 256) {
        const int j = i / KPAD, u = i % KPAD;
        float v = 0.0f;
        if (u < W_LEN) {
            if (j < RANK)                      v = K[(j * 2 + 0) * W_LEN + (W_LEN - 1 - u)];
            else if (j >= 16 && j < 16 + RANK) v = K[((j - 16) * 2 + 1) * W_LEN + (W_LEN - 1 - u)];
        }
        Kf[i] = v;
    }
}

// ---------------- pass 0: zero the stats accumulators ----------------
__global__ void zero_ws_kernel(float* __restrict__ p, int nelem) {
    const int i = blockIdx.x * blockDim.x + threadIdx.x;
    if (i < nelem) p[i] = 0.0f;
}

// ---------------- pass 1: per-(n,k) sum / sumsq of c ----------------
__global__ void __launch_bounds__(256)
pass1_stats(const float* __restrict__ X, const float* __restrict__ K,
            float* __restrict__ sumA, float* __restrict__ sumQ) {
    __shared__ float Kf[ROWS * KPAD];
    __shared__ float Xw[CH1 + KPAD + 8];
    const int tid  = threadIdx.x;
    const int lane = tid & 31;
    const int wave = tid >> 5;
    stage_filters(K, Kf, tid);

    const int T0 = blockIdx.x * CH1;
    const int n0 = blockIdx.y * 16;

    for (int nn = 0; nn < 16; ++nn) {
        const int n = n0 + nn;
        __syncthreads();
        for (int i = tid; i < CH1 + KPAD; i += 256) {
            const int t = T0 - 50 + i;
            Xw[i] = (t >= 0 && t < T_LEN) ? X[n * T_LEN + t] : 0.0f;
        }
        __syncthreads();

        float s[8], q[8];
#pragma unroll
        for (int r = 0; r < 8; ++r) { s[r] = 0.0f; q[r] = 0.0f; }

        for (int sub = 0; sub < NSUB1; ++sub) {
            const int toff = (wave * NSUB1 + sub) * 16;
            v8f ar, ai;
            conv_tile(Xw, toff, Kf, lane, ar, ai);
            const int  t    = T0 + toff + (lane & 15);
            const bool tval = (t < T_LEN);
#pragma unroll
            for (int r = 0; r < 8; ++r) {
                const float re = ar[r], im = ai[r];
                float c = sqrtf(re * re + im * im);
                c = tval ? c : 0.0f;
                s[r] += c; q[r] += c * c;
            }
        }
        // reduce over the 16 lanes of each half (same filter set per half)
#pragma unroll
        for (int r = 0; r < 8; ++r) {
            float sv = s[r], qv = q[r];
            for (int off = 1; off < 16; off <<= 1) {
                sv += __shfl_xor(sv, off, 32);
                qv += __shfl_xor(qv, off, 32);
            }
            if ((lane & 15) == 0) {
                const int k = (lane >> 4) * 8 + r;
                if (k < RANK) {
                    atomicAdd(&sumA[n * RANK + k], sv);
                    atomicAdd(&sumQ[n * RANK + k], qv);
                }
            }
        }
    }
}

// ---------------- pass 2: fold stats into readout weights ----------------
__global__ void finalize_stats(const float* __restrict__ B1,
                               const float* __restrict__ sumA,
                               const float* __restrict__ sumQ,
                               float* __restrict__ B1s, float* __restrict__ ybar) {
    const int tid = threadIdx.x;
    if (tid < RANK) ybar[tid] = 0.0f;
    __syncthreads();
    for (int i = tid; i < N_IN * RANK; i += blockDim.x) {
        const float s    = sumA[i], q = sumQ[i];
        const float mean = s / (float)T_LEN;
        const float var  = (q - s * mean) / (float)(T_LEN - 1);   // ddof = 1
        const float w    = B1[i] * rsqrtf(var);
        B1s[i] = w;
        atomicAdd(&ybar[i % RANK], w * mean);
    }
}

// ---------------- pass 3: conv + standardize-free readout ----------------
__global__ void __launch_bounds__(256)
pass3_out(const float* __restrict__ X, const float* __restrict__ K,
          const float* __restrict__ B2, const float* __restrict__ bias,
          const float* __restrict__ B1s, const float* __restrict__ ybar,
          float* __restrict__ out) {
    __shared__ float Kf[ROWS * KPAD];
    __shared__ float Xw[CH3 + KPAD + 8];
    __shared__ float B1sS[N_IN * RANK];
    __shared__ float B2S[RANK * N_OUT];
    __shared__ float ybS[RANK];
    __shared__ float biasS[N_OUT];
    const int tid  = threadIdx.x;
    const int lane = tid & 31;
    const int wave = tid >> 5;
    const int hi   = lane >> 4;

    stage_filters(K, Kf, tid);
    for (int i = tid; i < N_IN * RANK; i += 256) B1sS[i] = B1s[i];
    for (int i = tid; i < RANK * N_OUT; i += 256) B2S[i] = B2[i];
    if (tid < RANK)  ybS[tid]  = ybar[tid];
    if (tid < N_OUT) biasS[tid] = bias[tid];

    const int T0   = blockIdx.x * CH3;
    const int toff = wave * 16;
    const int t    = T0 + toff + (lane & 15);

    float y[8];
#pragma unroll
    for (int r = 0; r < 8; ++r) y[r] = 0.0f;

    for (int n = 0; n < N_IN; ++n) {
        __syncthreads();
        for (int i = tid; i < CH3 + KPAD; i += 256) {
            const int tt = T0 - 50 + i;
            Xw[i] = (tt >= 0 && tt < T_LEN) ? X[n * T_LEN + tt] : 0.0f;
        }
        __syncthreads();
        v8f ar, ai;
        conv_tile(Xw, toff, Kf, lane, ar, ai);
#pragma unroll
        for (int r = 0; r < 8; ++r) {
            const int   k  = hi * 8 + r;
            const float re = ar[r], im = ai[r];
            const float c  = sqrtf(re * re + im * im);
            const float w  = (k < RANK) ? B1sS[n * RANK + k] : 0.0f;
            y[r] += w * c;                      // y[k,t] = sum_n B1s[n,k]*c[n,k,t]
        }
    }

    // out[m,t] = sum_k B2[k,m]*(y[k,t]-ybar[k]) + bias[m]
    float yp[8];
    int   kc[8];
#pragma unroll
    for (int r = 0; r < 8; ++r) {
        const int k = hi * 8 + r;
        yp[r] = (k < RANK) ? (y[r] - ybS[k]) : 0.0f;
        kc[r] = (k < RANK) ? k : 0;            // clamped (yp==0 kills it)
    }
    for (int m = 0; m < N_OUT; ++m) {
        float p = 0.0f;
#pragma unroll
        for (int r = 0; r < 8; ++r) p += yp[r] * B2S[kc[r] * N_OUT + m];
        p += __shfl_xor(p, 16, 32);            // combine k=0..7 half with k=8..9 half
        if (hi == 0 && t < T_LEN) out[m * T_LEN + t] = p + biasS[m];
    }
}

// ---------------- launch ----------------
extern "C" void kernel_launch(void* const* d_in, const int* in_sizes, int n_in,
                              void* d_out, int out_size, void* d_ws, size_t ws_size,
                              hipStream_t stream) {
    const float* X    = (const float*)d_in[0];
    const float* K    = (const float*)d_in[1];
    const float* B1   = (const float*)d_in[2];
    const float* B2   = (const float*)d_in[3];
    const float* bias = (const float*)d_in[4];
    float*       out  = (float*)d_out;

    float* ws   = (float*)d_ws;
    float* sumA = ws;                    // 1280
    float* sumQ = ws + N_IN * RANK;      // 1280
    float* B1s  = ws + 2 * N_IN * RANK;  // 1280
    float* ybar = ws + 3 * N_IN * RANK;  // 10

    const int nstats = 2 * N_IN * RANK;
    zero_ws_kernel<<<(nstats + 255) / 256, 256, 0, stream>>>(ws, nstats);

    const int nchunk1 = (T_LEN + CH1 - 1) / CH1;   // 98
    pass1_stats<<<dim3(nchunk1, N_IN / 16), 256, 0, stream>>>(X, K, sumA, sumQ);

    finalize_stats<<<1, 256, 0, stream>>>(B1, sumA, sumQ, B1s, ybar);

    const int nchunk3 = (T_LEN + CH3 - 1) / CH3;   // 391
    pass3_out<<<nchunk3, 256, 0, stream>>>(X, K, B2, bias, B1s, ybar, out);
}